// ATGCNN_14259291423187
// MI455X (gfx1250) — compile-verified
//
#include <hip/hip_runtime.h>
#include <math.h>

// Problem constants (match reference setup_inputs)
#define NNODES 10000
#define NEDGES 160000
#define DIN    512
#define NHEAD  4
#define CH     128
#define HFINAL 2
#define CFINAL 512
#define NEG_SLOPE 0.2f

typedef float v2f __attribute__((ext_vector_type(2)));
typedef float v8f __attribute__((ext_vector_type(8)));

// ---------------------------------------------------------------------------
// utility: fill buffer with a constant (used for -inf segment-max init, zeros)
// ---------------------------------------------------------------------------
__global__ void fill_f32(float* __restrict__ p, float v, int n) {
  int i = blockIdx.x * blockDim.x + threadIdx.x;
  if (i < n) p[i] = v;
}

// ---------------------------------------------------------------------------
// GEMM: C[M,Nout] = A[M,K] @ B[K,Nout], fp32 via V_WMMA_F32_16X16X4_F32.
// Each wave computes a 16x32 output strip = TWO adjacent 16x16 tiles sharing
// one A fragment, with two independent accumulators so the WMMA dependency
// chains interleave on the matrix pipe (hides XDL latency) and A traffic is
// halved.  K stepped by 4 per WMMA.
// A-frag layout (32b 16x4): lanes 0-15 -> M=lane, v0=K0,v1=K1;
//                           lanes 16-31 -> M=lane-16, v0=K2,v1=K3.
// B-frag layout (4x16): N striped across lanes (symmetric).
// C/D layout: VGPR r: lanes 0-15 -> M=r, lanes 16-31 -> M=r+8; N=lane&15.
// Early-exit branch is wave-uniform so EXEC is all-ones at every WMMA.
// ---------------------------------------------------------------------------
__global__ void gemm_wmma_f32(const float* __restrict__ A,
                              const float* __restrict__ B,
                              float* __restrict__ C,
                              int M, int K, int Nout) {
  const int lane = threadIdx.x & 31;
  const int wave = threadIdx.x >> 5;
  const int pairs_n = Nout >> 5;                      // N-tile pairs
  const int pairs_total = (M >> 4) * pairs_n;
  const int pair_id = blockIdx.x * (blockDim.x >> 5) + wave;
  if (pair_id >= pairs_total) return;                 // wave-uniform

  const int tm  = pair_id / pairs_n;
  const int tp  = pair_id - tm * pairs_n;
  const int n0  = tp * 32;                            // first column of strip
  const int l15   = lane & 15;
  const int khalf = (lane >> 4) << 1;                 // 0 | 2

  const float* __restrict__ arow  = A + (size_t)(tm * 16 + l15) * K;
  const float* __restrict__ bcol0 = B + (n0 + l15);
  const float* __restrict__ bcol1 = B + (n0 + 16 + l15);

  v8f acc0 = {};
  v8f acc1 = {};
  #pragma unroll 4
  for (int k = 0; k < K; k += 4) {
    v2f av, bv0, bv1;
    const size_t r0 = (size_t)(k + khalf) * Nout;
    const size_t r1 = (size_t)(k + khalf + 1) * Nout;
    av.x  = arow[k + khalf];
    av.y  = arow[k + khalf + 1];
    bv0.x = bcol0[r0];
    bv0.y = bcol0[r1];
    bv1.x = bcol1[r0];
    bv1.y = bcol1[r1];
    // two independent accumulator chains -> interleaved matrix-pipe issue
    acc0 = __builtin_amdgcn_wmma_f32_16x16x4_f32(false, av, false, bv0,
                                                 (short)0, acc0, false, false);
    acc1 = __builtin_amdgcn_wmma_f32_16x16x4_f32(false, av, false, bv1,
                                                 (short)0, acc1, false, false);
  }

  const int crow = tm * 16 + ((lane >> 4) << 3);
  float* cp0 = C + (size_t)crow * Nout + (n0 + l15);
  float* cp1 = cp0 + 16;
  #pragma unroll
  for (int r = 0; r < 8; ++r) {
    cp0[(size_t)r * Nout] = acc0[r];
    cp1[(size_t)r * Nout] = acc1[r];
  }
}

// ---------------------------------------------------------------------------
// Attention scores: es[n,h] = sum_c h[n,h,c]*a_src[h,c]; ed likewise.
// One wave32 per (node, head); wave shuffle reduction.
// ---------------------------------------------------------------------------
__global__ void scores_kernel(const float* __restrict__ h,
                              const float* __restrict__ a_src,
                              const float* __restrict__ a_dst,
                              float* __restrict__ es, float* __restrict__ ed,
                              int Nn, int Hh, int Cc) {
  const int wid  = (blockIdx.x * blockDim.x + threadIdx.x) >> 5;  // wave-uniform
  const int lane = threadIdx.x & 31;
  if (wid >= Nn * Hh) return;
  const int n  = wid / Hh;
  const int hh = wid - n * Hh;
  const float* hp = h + ((size_t)n * Hh + hh) * Cc;
  const float* sp = a_src + hh * Cc;
  const float* dp = a_dst + hh * Cc;
  float s = 0.f, d = 0.f;
  for (int c = lane; c < Cc; c += 32) {
    float v = hp[c];
    s += v * sp[c];
    d += v * dp[c];
  }
  #pragma unroll
  for (int off = 16; off > 0; off >>= 1) {
    s += __shfl_down(s, off, 32);
    d += __shfl_down(d, off, 32);
  }
  if (lane == 0) { es[wid] = s; ed[wid] = d; }
}

// ---------------------------------------------------------------------------
// float atomic max via CAS (portable; init value must be -inf, stored values
// are always finite so the < compare is well-defined)
// ---------------------------------------------------------------------------
__device__ __forceinline__ void atomicMaxF32(float* addr, float val) {
  unsigned* ua = (unsigned*)addr;
  unsigned old = *ua;
  while (__uint_as_float(old) < val) {
    unsigned assumed = old;
    old = atomicCAS(ua, assumed, __float_as_uint(val));
    if (old == assumed) break;
  }
}

// e = leaky_relu(es[src]+ed[dst]); segment-max into m[dst]
__global__ void edge_logits(const int* __restrict__ src, const int* __restrict__ dst,
                            const float* __restrict__ es, const float* __restrict__ ed,
                            float* __restrict__ eatt, float* __restrict__ m,
                            int E_, int Hh) {
  int i = blockIdx.x * blockDim.x + threadIdx.x;
  if (i >= E_ * Hh) return;
  int e  = i / Hh;
  int hh = i - e * Hh;
  int s = src[e], d = dst[e];
  float v = es[s * Hh + hh] + ed[d * Hh + hh];
  v = (v > 0.f) ? v : NEG_SLOPE * v;
  eatt[i] = v;
  atomicMaxF32(&m[d * Hh + hh], v);
}

// ex = exp(e - m[dst]); segment-sum into den[dst]
__global__ void edge_exp(const int* __restrict__ dst,
                         float* __restrict__ eatt, const float* __restrict__ m,
                         float* __restrict__ den, int E_, int Hh) {
  int i = blockIdx.x * blockDim.x + threadIdx.x;
  if (i >= E_ * Hh) return;
  int e  = i / Hh;
  int hh = i - e * Hh;
  int d = dst[e];
  float ex = expf(eatt[i] - m[d * Hh + hh]);
  eatt[i] = ex;
  atomicAdd(&den[d * Hh + hh], ex);
}

// agg[dst] += h[src] * alpha   (one workgroup per edge, coalesced channels)
__global__ void edge_aggregate(const int* __restrict__ src, const int* __restrict__ dst,
                               const float* __restrict__ h,
                               const float* __restrict__ eatt,
                               const float* __restrict__ den,
                               float* __restrict__ agg,
                               int E_, int Hh, int Cc) {
  const int e = blockIdx.x;
  const int s = src[e], d = dst[e];
  const int F = Hh * Cc;
  const float* __restrict__ hs = h + (size_t)s * F;
  float* __restrict__ ad = agg + (size_t)d * F;
  for (int j = threadIdx.x; j < F; j += blockDim.x) {
    const int hh = j / Cc;
    const float alpha = eatt[(size_t)e * Hh + hh] / (den[d * Hh + hh] + 1e-16f);
    atomicAdd(&ad[j], hs[j] * alpha);
  }
}

// x_out = elu(agg + bias + x_prev)   (layers 1 and 2; concat heads)
__global__ void combine_elu(const float* __restrict__ agg, const float* __restrict__ bias,
                            const float* __restrict__ xprev, float* __restrict__ xout,
                            int Nn, int F) {
  int i = blockIdx.x * blockDim.x + threadIdx.x;
  if (i >= Nn * F) return;
  float v = agg[i] + bias[i % F] + xprev[i];
  xout[i] = (v > 0.f) ? v : (expf(v) - 1.f);
}

// x_out = mean_heads(agg) + bias + x_prev   (final layer, no activation)
__global__ void combine_mean(const float* __restrict__ agg, const float* __restrict__ bias,
                             const float* __restrict__ xprev, float* __restrict__ xout,
                             int Nn, int Cc) {
  int i = blockIdx.x * blockDim.x + threadIdx.x;
  if (i >= Nn * Cc) return;
  int n = i / Cc;
  int c = i - n * Cc;
  float v = 0.5f * (agg[(size_t)n * 2 * Cc + c] + agg[(size_t)n * 2 * Cc + Cc + c]);
  xout[i] = v + bias[c] + xprev[i];
}

// ---------------------------------------------------------------------------
// Host-side: one GAT layer
// ---------------------------------------------------------------------------
static void launch_gat_layer(const float* x_in, int Din_,
                             const float* W, const float* asrc, const float* adst,
                             const float* bias, int Hh, int Cc,
                             const int* src, const int* dst,
                             float* h, float* es, float* ed, float* m, float* den,
                             float* eatt, float* agg, float* x_out,
                             int final_mean, hipStream_t stream) {
  const int F  = Hh * Cc;
  const int NH = NNODES * Hh;
  const int NF = NNODES * F;
  const int EH = NEDGES * Hh;

  fill_f32<<<(NH + 255) / 256, 256, 0, stream>>>(m,  -INFINITY, NH);
  fill_f32<<<(NH + 255) / 256, 256, 0, stream>>>(den, 0.f,      NH);
  fill_f32<<<(NF + 255) / 256, 256, 0, stream>>>(agg, 0.f,      NF);

  const int pairs = (NNODES / 16) * (F / 32);   // 16x32 strips per wave
  gemm_wmma_f32<<<(pairs + 3) / 4, 128, 0, stream>>>(x_in, W, h, NNODES, Din_, F);

  scores_kernel<<<(NH * 32 + 255) / 256, 256, 0, stream>>>(h, asrc, adst, es, ed,
                                                           NNODES, Hh, Cc);

  edge_logits<<<(EH + 255) / 256, 256, 0, stream>>>(src, dst, es, ed, eatt, m,
                                                    NEDGES, Hh);
  edge_exp<<<(EH + 255) / 256, 256, 0, stream>>>(dst, eatt, m, den, NEDGES, Hh);

  edge_aggregate<<<NEDGES, 128, 0, stream>>>(src, dst, h, eatt, den, agg,
                                             NEDGES, Hh, Cc);

  if (!final_mean) {
    combine_elu<<<(NF + 255) / 256, 256, 0, stream>>>(agg, bias, x_in, x_out,
                                                      NNODES, F);
  } else {
    const int NC = NNODES * Cc;
    combine_mean<<<(NC + 255) / 256, 256, 0, stream>>>(agg, bias, x_in, x_out,
                                                       NNODES, Cc);
  }
}

extern "C" void kernel_launch(void* const* d_in, const int* in_sizes, int n_in,
                              void* d_out, int out_size, void* d_ws, size_t ws_size,
                              hipStream_t stream) {
  const float* graph = (const float*)d_in[0];
  const int*   ei    = (const int*)d_in[1];
  const int*   src   = ei;            // edge_index[0]
  const int*   dst   = ei + NEDGES;   // edge_index[1]
  const float* W1  = (const float*)d_in[2];
  const float* as1 = (const float*)d_in[3];
  const float* ad1 = (const float*)d_in[4];
  const float* b1  = (const float*)d_in[5];
  const float* W2  = (const float*)d_in[6];
  const float* as2 = (const float*)d_in[7];
  const float* ad2 = (const float*)d_in[8];
  const float* b2  = (const float*)d_in[9];
  const float* W3  = (const float*)d_in[10];
  const float* as3 = (const float*)d_in[11];
  const float* ad3 = (const float*)d_in[12];
  const float* b3  = (const float*)d_in[13];

  // workspace partition (floats); total ~126 MB
  float* ws = (float*)d_ws;
  size_t off = 0;
  float* xA   = ws + off; off += (size_t)NNODES * 512;   // layer-1 output
  float* xB   = ws + off; off += (size_t)NNODES * 512;   // layer-2 output
  float* hbuf = ws + off; off += (size_t)NNODES * 1024;  // x@W (max layer-3 width)
  float* agg  = ws + off; off += (size_t)NNODES * 1024;  // segment-sum accumulator
  float* es   = ws + off; off += (size_t)NNODES * NHEAD;
  float* ed   = ws + off; off += (size_t)NNODES * NHEAD;
  float* m    = ws + off; off += (size_t)NNODES * NHEAD;
  float* den  = ws + off; off += (size_t)NNODES * NHEAD;
  float* eatt = ws + off; off += (size_t)NEDGES * NHEAD;

  // Layer 1: graph -> xA   (4 heads x 128, concat, ELU + residual)
  launch_gat_layer(graph, DIN, W1, as1, ad1, b1, NHEAD, CH, src, dst,
                   hbuf, es, ed, m, den, eatt, agg, xA, 0, stream);
  // Layer 2: xA -> xB      (4 heads x 128, concat, ELU + residual)
  launch_gat_layer(xA, NHEAD * CH, W2, as2, ad2, b2, NHEAD, CH, src, dst,
                   hbuf, es, ed, m, den, eatt, agg, xB, 0, stream);
  // Layer 3: xB -> d_out   (2 heads x 512, head-mean, residual, no ELU)
  launch_gat_layer(xB, NHEAD * CH, W3, as3, ad3, b3, HFINAL, CFINAL, src, dst,
                   hbuf, es, ed, m, den, eatt, agg, (float*)d_out, 1, stream);
}